// UnfoldFoldModule_13451837571944
// MI455X (gfx1250) — compile-verified
//
#include <hip/hip_runtime.h>
#include <stdint.h>

// out[b,c,y,x] = x[b,c,y,x] * ((y%3!=0) && (x%3!=0))   (fold(unfold(x)) == coverage-count mask, count in {0,1})
// Pure bandwidth-bound streaming kernel; uses gfx1250 async global->LDS loads + ASYNCcnt.

typedef int v4i __attribute__((ext_vector_type(4)));
typedef __attribute__((address_space(1))) v4i GV4;  // global 128-bit payload
typedef __attribute__((address_space(3))) v4i LV4;  // LDS 128-bit payload

__device__ __forceinline__ void async_load_b128(const void* gsrc, unsigned lds_off) {
#if __has_builtin(__builtin_amdgcn_global_load_async_to_lds_b128)
    __builtin_amdgcn_global_load_async_to_lds_b128(
        (GV4*)(uintptr_t)gsrc,                 // AS1 v4i* (global VA == generic VA, inttoptr)
        (LV4*)(uintptr_t)(uint64_t)lds_off,    // AS3 v4i* from LDS byte offset (inttoptr)
        0, 0);
#else
    unsigned long long ga = (unsigned long long)(uintptr_t)gsrc;
    asm volatile("global_load_async_to_lds_b128 %0, %1, off"
                 :: "v"(lds_off), "v"(ga)
                 : "memory");
#endif
}

__device__ __forceinline__ void wait_async0() {
#if __has_builtin(__builtin_amdgcn_s_wait_asynccnt)
    __builtin_amdgcn_s_wait_asynccnt(0);
#else
    asm volatile("s_wait_asynccnt 0x0" ::: "memory");
#endif
}

__global__ __launch_bounds__(256) void unfold_fold_mask_kernel(
    const float4* __restrict__ x, float4* __restrict__ out, int n4)
{
    __shared__ float4 buf[256];
    const int tid = (int)threadIdx.x;
    const int t   = (int)blockIdx.x * 256 + tid;
    if (t >= n4) return;

    // Tensor is (B,C,32,32) row-major; 8 float4 per 32-wide row.
    const unsigned w4 = (unsigned)t & 7u;          // which float4 in the row -> w0 = 4*w4
    const unsigned h  = ((unsigned)t >> 3) & 31u;  // row within image
    const bool rowLive = (h % 3u) != 0u;           // h%3==0 rows are fully zero: skip the load

    const unsigned lds_off = (unsigned)(uintptr_t)&buf[tid]; // low 32 bits of generic LDS ptr = LDS byte offset

    if (rowLive) {
        async_load_b128(x + t, lds_off);           // global -> LDS, tracked by ASYNCcnt
    }
    wait_async0();                                 // wave-level: wait for our in-flight async copy
    asm volatile("" ::: "memory");                 // compiler barrier: don't hoist the LDS read

    float4 v = make_float4(0.f, 0.f, 0.f, 0.f);
    if (rowLive) {
        v = buf[tid];                              // ds_load_b128 of our own slot (no cross-lane: no barrier)
        // Component j is zeroed iff (4*w4 + j) % 3 == 0  <=>  (w4 + j) % 3 == 0
        const unsigned r = w4 % 3u;
        if (r == 0u)      { v.x = 0.f; v.w = 0.f; }
        else if (r == 1u) { v.z = 0.f; }
        else              { v.y = 0.f; }
    }
    out[t] = v;                                    // coalesced global_store_b128 (full overwrite of d_out)
}

extern "C" void kernel_launch(void* const* d_in, const int* in_sizes, int n_in,
                              void* d_out, int out_size, void* d_ws, size_t ws_size,
                              hipStream_t stream) {
    const float4* x   = (const float4*)d_in[0];
    float4*       out = (float4*)d_out;
    const int n4 = in_sizes[0] >> 2;               // 33,554,432 floats -> 8,388,608 float4
    const int block = 256;
    const int grid  = (n4 + block - 1) / block;    // 32768 blocks
    unfold_fold_mask_kernel<<<grid, block, 0, stream>>>(x, out, n4);
}